// RGCN_5789615915676
// MI455X (gfx1250) — compile-verified
//
#include <hip/hip_runtime.h>

// Problem constants (match reference)
constexpr int NN = 50000;   // nodes
constexpr int EE = 250000;  // edges per relation per layer
constexpr int DD = 256;     // feature dim
constexpr int CC = 349;     // classes
constexpr int LL = 2;       // layers
constexpr int RR = 3;       // relations

typedef float v2f __attribute__((ext_vector_type(2)));
typedef float v8f __attribute__((ext_vector_type(8)));

// ---------- CDNA5 async global->LDS staging (guarded, with sync fallback) ----------
#if defined(__has_builtin)
#if __has_builtin(__builtin_amdgcn_global_load_async_to_lds_b128) && \
    __has_builtin(__builtin_amdgcn_global_load_async_to_lds_b32)  && \
    __has_builtin(__builtin_amdgcn_s_wait_asynccnt)
#define HAVE_ASYNC_LDS 1
#endif
#endif
#ifndef HAVE_ASYNC_LDS
#define HAVE_ASYNC_LDS 0
#endif

#if HAVE_ASYNC_LDS
// Builtin signatures (from hipcc diagnostics):
//   b128: (AS1 int4* src, AS3 int4* dst, imm offset, imm cpol)
//   b32 : (AS1 int*  src, AS3 int*  dst, imm offset, imm cpol)
typedef int v4i_t __attribute__((vector_size(16)));
typedef __attribute__((address_space(1))) v4i_t* g128p_t;
typedef __attribute__((address_space(3))) v4i_t* l128p_t;
typedef __attribute__((address_space(1))) int*   g32p_t;
typedef __attribute__((address_space(3))) int*   l32p_t;
__device__ __forceinline__ void cp_async16(const float* gp, float* lp) {
    __builtin_amdgcn_global_load_async_to_lds_b128((g128p_t)gp, (l128p_t)lp, 0, 0);
}
__device__ __forceinline__ void cp_async4(const float* gp, float* lp) {
    __builtin_amdgcn_global_load_async_to_lds_b32((g32p_t)gp, (l32p_t)lp, 0, 0);
}
#define ASYNC_WAIT() __builtin_amdgcn_s_wait_asynccnt(0)
#else
#define ASYNC_WAIT()
#endif

// -------------------- utility kernels --------------------

__global__ __launch_bounds__(256) void zero4_kernel(float4* p, long n4) {
    long i = (long)blockIdx.x * blockDim.x + threadIdx.x;
    if (i < n4) p[i] = make_float4(0.f, 0.f, 0.f, 0.f);
}

__global__ __launch_bounds__(256) void count_deg_kernel(const int* __restrict__ edges,
                                                        float* __restrict__ deg_out,
                                                        float* __restrict__ deg_in) {
    long i = (long)blockIdx.x * blockDim.x + threadIdx.x;
    if (i >= (long)LL * RR * EE) return;
    int e  = (int)(i % EE);
    int lr = (int)(i / EE);          // l*R + r
    long base = (long)lr * 2 * EE;
    int src = edges[base + e];
    int dst = edges[base + EE + e];
    atomicAdd(&deg_out[(long)lr * NN + src], 1.0f);
    atomicAdd(&deg_in [(long)lr * NN + dst], 1.0f);
}

__global__ __launch_bounds__(256) void finalize_deg_kernel(float* __restrict__ p, long n) {
    long i = (long)blockIdx.x * blockDim.x + threadIdx.x;
    if (i < n) p[i] = rsqrtf(fmaxf(p[i], 1.0f));
}

// One wave per edge: agg[r][dst][:] += h[src][:] * rs_out[r][src]
__global__ __launch_bounds__(256) void scatter_kernel(const float* __restrict__ h,
                                                      const int* __restrict__ edges_l,   // [R][2][E]
                                                      const float* __restrict__ rsout_l, // [R][N]
                                                      float* __restrict__ agg) {         // [R][N][D]
    int widx = threadIdx.x >> 5;
    int lane = threadIdx.x & 31;
    long wg = (long)blockIdx.x * 8 + widx;
    if (wg >= (long)RR * EE) return;
    int r = (int)(wg / EE);
    int e = (int)(wg % EE);
    long base = (long)r * 2 * EE;
    int src = edges_l[base + e];
    int dst = edges_l[base + EE + e];
    float s = rsout_l[(long)r * NN + src];
    const float4* hp = (const float4*)(h + (long)src * DD);
    float* ap = agg + ((long)r * NN + dst) * DD;
#pragma unroll
    for (int j = lane; j < DD / 4; j += 32) {
        float4 v = hp[j];
        atomicAdd(&ap[4 * j + 0], v.x * s);
        atomicAdd(&ap[4 * j + 1], v.y * s);
        atomicAdd(&ap[4 * j + 2], v.z * s);
        atomicAdd(&ap[4 * j + 3], v.w * s);
    }
}

// -------------------- WMMA GEMM tiling parameters --------------------
// Block tile 128x64, 8 wave32s as 4x2 grid of 32x32 wave tiles (2x2 WMMA register
// blocking), K staged in chunks of 32 through double-buffered LDS.
#define KC 32
#define APAD 36   // 32 + 4 pad: rows stay 16B-aligned, LDS banks conflict-free

__device__ __forceinline__ v8f wmma_f32(v2f a, v2f b, v8f c) {
    return __builtin_amdgcn_wmma_f32_16x16x4_f32(false, a, false, b, (short)0, c, false, false);
}

// ---- staging helpers (async if available, sync fallback otherwise) ----

__device__ __forceinline__ void stage_layer_chunk(const float* __restrict__ agg,
                                                  const float* __restrict__ Wl,
                                                  float (*As)[APAD], float (*Bs)[64],
                                                  int m0, int n0, int kb, int tid) {
    const int r  = kb >> 8;    // relation
    const int kk = kb & 255;   // k within relation
    // A: 128 x 32 (pure copy; per-row scale applied later in registers)
#pragma unroll
    for (int f = tid; f < 1024; f += 256) {
        int row = f >> 3;
        int c4  = (f & 7) << 2;
        int grow = m0 + row;
        if (grow >= NN) grow = NN - 1;               // clamp: no OOB fault; scale==0 kills it
        const float* gp = agg + ((long)r * NN + grow) * DD + kk + c4;
        float* lp = &As[row][c4];
#if HAVE_ASYNC_LDS
        cp_async16(gp, lp);
#else
        float4 v = *(const float4*)gp;
        lp[0] = v.x; lp[1] = v.y; lp[2] = v.z; lp[3] = v.w;
#endif
    }
    // B: 32 x 64 from W[r]
#pragma unroll
    for (int f = tid; f < 512; f += 256) {
        int row = f >> 4;
        int c4  = (f & 15) << 2;
        const float* gp = Wl + ((long)r * DD + kk + row) * DD + n0 + c4;
        float* lp = &Bs[row][c4];
#if HAVE_ASYNC_LDS
        cp_async16(gp, lp);
#else
        *(float4*)lp = *(const float4*)gp;
#endif
    }
}

__device__ __forceinline__ void stage_out_chunk(const float* __restrict__ h,
                                                const float* __restrict__ Wout,
                                                float (*As)[APAD], float (*Bs)[64],
                                                int m0, int n0, int kk, int tid) {
    // A: 128 x 32 from h
#pragma unroll
    for (int f = tid; f < 1024; f += 256) {
        int row = f >> 3;
        int c4  = (f & 7) << 2;
        int grow = m0 + row;
        if (grow >= NN) grow = NN - 1;               // clamped rows only feed unstored outputs
        const float* gp = h + (long)grow * DD + kk + c4;
        float* lp = &As[row][c4];
#if HAVE_ASYNC_LDS
        cp_async16(gp, lp);
#else
        float4 v = *(const float4*)gp;
        lp[0] = v.x; lp[1] = v.y; lp[2] = v.z; lp[3] = v.w;
#endif
    }
    // B: 32 x 64 from Wout (rows of length 349 -> scalar, column-clamped)
#pragma unroll
    for (int f = tid; f < 2048; f += 256) {
        int row = f >> 6;
        int col = f & 63;
        int gc  = n0 + col;
        int gcc = gc < CC ? gc : CC - 1;             // clamped cols only feed unstored outputs
        const float* gp = Wout + (long)(kk + row) * CC + gcc;
        float* lp = &Bs[row][col];
#if HAVE_ASYNC_LDS
        cp_async4(gp, lp);
#else
        *lp = *gp;
#endif
    }
}

// -------------------- WMMA GEMM: layer update --------------------
// hout[n][j] = act( (1/3)*( sum_r (agg[r][n][:]*rin[r][n]) @ W[r][:][j] + sum_r b[r][j] ) )
__global__ __launch_bounds__(256) void gemm_layer_kernel(const float* __restrict__ agg,  // [3][N][D]
                                                         const float* __restrict__ rin,  // [3][N]
                                                         const float* __restrict__ Wl,   // [3][D][D]
                                                         const float* __restrict__ bl,   // [3][D]
                                                         float* __restrict__ hout,       // [N][D]
                                                         int relu) {
    __shared__ __align__(16) float As[2][128][APAD];
    __shared__ __align__(16) float Bs[2][KC][64];

    const int tid  = threadIdx.x;
    const int lane = tid & 31;
    const int w    = tid >> 5;
    const int g    = lane >> 4;
    const int ln   = lane & 15;
    const int m0   = blockIdx.x * 128;
    const int n0   = blockIdx.y * 64;
    const int mw   = (w >> 1) << 5;   // wave tile row base: 0/32/64/96
    const int nw   = (w & 1) << 5;    // wave tile col base: 0/32

    v8f c00 = {0,0,0,0,0,0,0,0}, c01 = {0,0,0,0,0,0,0,0};
    v8f c10 = {0,0,0,0,0,0,0,0}, c11 = {0,0,0,0,0,0,0,0};

    const int mA0 = m0 + mw + ln;
    const int mA1 = mA0 + 16;

    constexpr int NCH = (RR * DD) / KC;   // 24 chunks
    stage_layer_chunk(agg, Wl, As[0], Bs[0], m0, n0, 0, tid);
    ASYNC_WAIT();
    __syncthreads();

    for (int i = 0; i < NCH; ++i) {
        const int cb = i & 1;
        if (i + 1 < NCH)   // prefetch next chunk into the other buffer
            stage_layer_chunk(agg, Wl, As[cb ^ 1], Bs[cb ^ 1], m0, n0, (i + 1) * KC, tid);

        const int r = (i * KC) >> 8;
        float s0 = (mA0 < NN) ? rin[(long)r * NN + mA0] : 0.f;
        float s1 = (mA1 < NN) ? rin[(long)r * NN + mA1] : 0.f;

#pragma unroll
        for (int k4 = 0; k4 < KC; k4 += 4) {
            const int k = k4 + 2 * g;
            v2f a0, a1, b0, b1;
            a0.x = As[cb][mw + ln][k];      a0.y = As[cb][mw + ln][k + 1];
            a1.x = As[cb][mw + 16 + ln][k]; a1.y = As[cb][mw + 16 + ln][k + 1];
            b0.x = Bs[cb][k][nw + ln];      b0.y = Bs[cb][k + 1][nw + ln];
            b1.x = Bs[cb][k][nw + 16 + ln]; b1.y = Bs[cb][k + 1][nw + 16 + ln];
            a0 *= s0;
            a1 *= s1;
            c00 = wmma_f32(a0, b0, c00);
            c01 = wmma_f32(a0, b1, c01);
            c10 = wmma_f32(a1, b0, c10);
            c11 = wmma_f32(a1, b1, c11);
        }
        ASYNC_WAIT();
        __syncthreads();
    }

    // epilogue: /3, + mean bias, optional relu
    const int col0 = n0 + nw + ln;
    const int col1 = col0 + 16;
    float bias0 = (bl[col0] + bl[DD + col0] + bl[2 * DD + col0]) * (1.0f / 3.0f);
    float bias1 = (bl[col1] + bl[DD + col1] + bl[2 * DD + col1]) * (1.0f / 3.0f);
#pragma unroll
    for (int i = 0; i < 8; ++i) {
        int r0 = m0 + mw + g * 8 + i;
        int r1 = r0 + 16;
        if (r0 < NN) {
            float v0 = c00[i] * (1.0f / 3.0f) + bias0;
            float v1 = c01[i] * (1.0f / 3.0f) + bias1;
            if (relu) { v0 = fmaxf(v0, 0.f); v1 = fmaxf(v1, 0.f); }
            hout[(long)r0 * DD + col0] = v0;
            hout[(long)r0 * DD + col1] = v1;
        }
        if (r1 < NN) {
            float v0 = c10[i] * (1.0f / 3.0f) + bias0;
            float v1 = c11[i] * (1.0f / 3.0f) + bias1;
            if (relu) { v0 = fmaxf(v0, 0.f); v1 = fmaxf(v1, 0.f); }
            hout[(long)r1 * DD + col0] = v0;
            hout[(long)r1 * DD + col1] = v1;
        }
    }
}

// -------------------- WMMA GEMM: output classifier --------------------
// out[n][c] = h[n][:] @ Wout[:][c] + bout[c]   (N x 256 x 349)
__global__ __launch_bounds__(256) void gemm_out_kernel(const float* __restrict__ h,     // [N][D]
                                                       const float* __restrict__ Wout,  // [D][C]
                                                       const float* __restrict__ bout,  // [C]
                                                       float* __restrict__ out) {       // [N][C]
    __shared__ __align__(16) float As[2][128][APAD];
    __shared__ __align__(16) float Bs[2][KC][64];

    const int tid  = threadIdx.x;
    const int lane = tid & 31;
    const int w    = tid >> 5;
    const int g    = lane >> 4;
    const int ln   = lane & 15;
    const int m0   = blockIdx.x * 128;
    const int n0   = blockIdx.y * 64;
    const int mw   = (w >> 1) << 5;
    const int nw   = (w & 1) << 5;

    v8f c00 = {0,0,0,0,0,0,0,0}, c01 = {0,0,0,0,0,0,0,0};
    v8f c10 = {0,0,0,0,0,0,0,0}, c11 = {0,0,0,0,0,0,0,0};

    constexpr int NCH = DD / KC;   // 8 chunks
    stage_out_chunk(h, Wout, As[0], Bs[0], m0, n0, 0, tid);
    ASYNC_WAIT();
    __syncthreads();

    for (int i = 0; i < NCH; ++i) {
        const int cb = i & 1;
        if (i + 1 < NCH)
            stage_out_chunk(h, Wout, As[cb ^ 1], Bs[cb ^ 1], m0, n0, (i + 1) * KC, tid);

#pragma unroll
        for (int k4 = 0; k4 < KC; k4 += 4) {
            const int k = k4 + 2 * g;
            v2f a0, a1, b0, b1;
            a0.x = As[cb][mw + ln][k];      a0.y = As[cb][mw + ln][k + 1];
            a1.x = As[cb][mw + 16 + ln][k]; a1.y = As[cb][mw + 16 + ln][k + 1];
            b0.x = Bs[cb][k][nw + ln];      b0.y = Bs[cb][k + 1][nw + ln];
            b1.x = Bs[cb][k][nw + 16 + ln]; b1.y = Bs[cb][k + 1][nw + 16 + ln];
            c00 = wmma_f32(a0, b0, c00);
            c01 = wmma_f32(a0, b1, c01);
            c10 = wmma_f32(a1, b0, c10);
            c11 = wmma_f32(a1, b1, c11);
        }
        ASYNC_WAIT();
        __syncthreads();
    }

    const int col0 = n0 + nw + ln;
    const int col1 = col0 + 16;
    float bias0 = (col0 < CC) ? bout[col0] : 0.f;
    float bias1 = (col1 < CC) ? bout[col1] : 0.f;
#pragma unroll
    for (int i = 0; i < 8; ++i) {
        int r0 = m0 + mw + g * 8 + i;
        int r1 = r0 + 16;
        if (r0 < NN) {
            if (col0 < CC) out[(long)r0 * CC + col0] = c00[i] + bias0;
            if (col1 < CC) out[(long)r0 * CC + col1] = c01[i] + bias1;
        }
        if (r1 < NN) {
            if (col0 < CC) out[(long)r1 * CC + col0] = c10[i] + bias0;
            if (col1 < CC) out[(long)r1 * CC + col1] = c11[i] + bias1;
        }
    }
}

// -------------------- host orchestration --------------------

extern "C" void kernel_launch(void* const* d_in, const int* in_sizes, int n_in,
                              void* d_out, int out_size, void* d_ws, size_t ws_size,
                              hipStream_t stream) {
    const float* x    = (const float*)d_in[0];
    const float* W    = (const float*)d_in[1];   // [L][R][D][D]
    const float* b    = (const float*)d_in[2];   // [L][R][D]
    const float* Wout = (const float*)d_in[3];   // [D][C]
    const float* bout = (const float*)d_in[4];   // [C]
    const int*  edges = (const int*)d_in[5];     // [L][R][2][E]
    float* out = (float*)d_out;

    float* ws      = (float*)d_ws;
    const long nLRN = (long)LL * RR * NN;
    float* rs_out = ws;                        // [L*R*N]  rsqrt(out_deg)
    float* rs_in  = ws + nLRN;                 // [L*R*N]  rsqrt(in_deg)
    float* agg    = rs_in + nLRN;              // [R][N][D]
    float* hbuf   = agg + (long)RR * NN * DD;  // [N][D]

    // 1) degrees
    long ndeg = 2 * nLRN;                      // divisible by 4
    zero4_kernel<<<(unsigned)((ndeg / 4 + 255) / 256), 256, 0, stream>>>((float4*)rs_out, ndeg / 4);
    long ncd = (long)LL * RR * EE;
    count_deg_kernel<<<(unsigned)((ncd + 255) / 256), 256, 0, stream>>>(edges, rs_out, rs_in);
    finalize_deg_kernel<<<(unsigned)((ndeg + 255) / 256), 256, 0, stream>>>(rs_out, ndeg);

    // 2) layers
    const long nagg = (long)RR * NN * DD;
    dim3 gemm1_grid((NN + 127) / 128, DD / 64);
    for (int l = 0; l < LL; ++l) {
        zero4_kernel<<<(unsigned)((nagg / 4 + 255) / 256), 256, 0, stream>>>((float4*)agg, nagg / 4);
        const float* hcur = (l == 0) ? x : hbuf;
        scatter_kernel<<<(unsigned)(((long)RR * EE) / 8), 256, 0, stream>>>(
            hcur, edges + (long)l * RR * 2 * EE, rs_out + (long)l * RR * NN, agg);
        gemm_layer_kernel<<<gemm1_grid, 256, 0, stream>>>(
            agg, rs_in + (long)l * RR * NN,
            W + (long)l * RR * DD * DD, b + (long)l * RR * DD,
            hbuf, (l == 0) ? 1 : 0);
    }

    // 3) classifier
    dim3 gemm2_grid((NN + 127) / 128, (CC + 63) / 64);
    gemm_out_kernel<<<gemm2_grid, 256, 0, stream>>>(hbuf, Wout, bout, out);
}